// RobotTrustModel_61658550501510
// MI455X (gfx1250) — compile-verified
//
#include <hip/hip_runtime.h>
#include <math.h>

typedef __attribute__((ext_vector_type(4))) float v4f;

__device__ __forceinline__ float softplus_f(float x) {
    // Numerically stable softplus: max(x,0) + log1p(exp(-|x|))
    return fmaxf(x, 0.0f) + log1pf(expf(-fabsf(x)));
}

__device__ __forceinline__ float sigmoid_f(float x) {
    return 1.0f / (1.0f + expf(-x));
}

__device__ __forceinline__ float piecewise_elem(float p, float l, float u, float inv_ul) {
    // ramp = (u - p) / (u - l + 1e-4); 1 if p <= l; 0 if p > u
    float r = (u - p) * inv_ul;
    r = (p >  u) ? 0.0f : r;
    r = (p <= l) ? 1.0f : r;
    return r;
}

__device__ __forceinline__ float smooth_elem(float p, float l, float u, float b, float inv_bul) {
    // 1 - (softplus(b(p-l)) - softplus(b(p-u))) / (b(u-l))
    return 1.0f - (softplus_f(b * (p - l)) - softplus_f(b * (p - u))) * inv_bul;
}

__global__ __launch_bounds__(256) void trust_curve_kernel(
    const float* __restrict__ bins,        // n floats
    float* __restrict__ out,               // n floats
    int n4,                                // n / 4 (float4 vectors)
    int tail,                              // n % 4
    const float* __restrict__ pre_beta,
    const float* __restrict__ pre_ll,
    const float* __restrict__ pre_lu)
{
    const int i = blockIdx.x * blockDim.x + threadIdx.x;

    // Uniform scalar parameters (1-element arrays); branch below is wave-uniform.
    const float pb = pre_beta[0];
    const float b  = pb * pb;
    const float l  = sigmoid_f(pre_ll[0]);
    const float u  = sigmoid_f(pre_lu[0]);
    const bool  use_smooth = (b < -50.0f);
    const float inv_ul  = 1.0f / (u - l + 1e-4f);
    const float inv_bul = use_smooth ? (1.0f / (b * (u - l))) : 0.0f;

    if (i < n4) {
        // Main body: 128-bit non-temporal streaming load/compute/store.
        const v4f* __restrict__ vin  = (const v4f*)bins;
        v4f*       __restrict__ vout = (v4f*)out;
        v4f p = __builtin_nontemporal_load(vin + i);
        v4f r;
        if (use_smooth) {
            r.x = smooth_elem(p.x, l, u, b, inv_bul);
            r.y = smooth_elem(p.y, l, u, b, inv_bul);
            r.z = smooth_elem(p.z, l, u, b, inv_bul);
            r.w = smooth_elem(p.w, l, u, b, inv_bul);
        } else {
            r.x = piecewise_elem(p.x, l, u, inv_ul);
            r.y = piecewise_elem(p.y, l, u, inv_ul);
            r.z = piecewise_elem(p.z, l, u, inv_ul);
            r.w = piecewise_elem(p.w, l, u, inv_ul);
        }
        __builtin_nontemporal_store(r, vout + i);
    } else if (i < n4 + tail) {
        // Tail (< 4 elements): scalar path.
        const int j = 4 * n4 + (i - n4);
        float p = bins[j];
        float r = use_smooth ? smooth_elem(p, l, u, b, inv_bul)
                             : piecewise_elem(p, l, u, inv_ul);
        __builtin_nontemporal_store(r, out + j);
    }
}

extern "C" void kernel_launch(void* const* d_in, const int* in_sizes, int n_in,
                              void* d_out, int out_size, void* d_ws, size_t ws_size,
                              hipStream_t stream) {
    const float* bins     = (const float*)d_in[0];
    const float* pre_beta = (const float*)d_in[1];
    const float* pre_ll   = (const float*)d_in[2];
    const float* pre_lu   = (const float*)d_in[3];
    float* out = (float*)d_out;

    const int n    = in_sizes[0];
    const int n4   = n >> 2;
    const int tail = n - (n4 << 2);

    const int threads = 256;                       // 8 wave32 per block
    const int total   = n4 + tail;                 // one thread per float4 + tail threads
    const int blocks  = (total + threads - 1) / threads;

    trust_curve_kernel<<<blocks, threads, 0, stream>>>(
        bins, out, n4, tail, pre_beta, pre_ll, pre_lu);
}